// SpGAT_609885356263
// MI455X (gfx1250) — compile-verified
//
#include <hip/hip_runtime.h>
#include <hip/hip_bf16.h>
#include <math.h>

// ---------------------------------------------------------------------------
// SpGAT on gfx1250 (MI455X).  fp32 end-to-end; dense GEMMs run on the CDNA5
// matrix pipe via V_WMMA_F32_16X16X4_F32 (wave32, 16x16 f32 tiles, K=4).
// Register-blocked: each wave owns a 64(M) x 64(N) strip -> 16 accumulators,
// so every B fragment load feeds 4 WMMAs (16 FLOP/byte in the hot loop).
// ---------------------------------------------------------------------------

#define F_IN   256
#define D_HID  64
#define HCAT   512          // H * D_HID
#define NHEAD  8
#define NCLS   10
#define NCLSP  16           // padded class dim for WMMA tile
#define GAT_ALPHA 0.2f

typedef float v2f __attribute__((ext_vector_type(2)));
typedef float v8f __attribute__((ext_vector_type(8)));

static __device__ __forceinline__ v8f wmma_f32(v2f a, v2f b, v8f c) {
    return __builtin_amdgcn_wmma_f32_16x16x4_f32(false, a, false, b, (short)0, c,
                                                 false, false);
}

// ---------------------------------------------------------------------------
// GEMM1:  hfeat[n][head*64+d] = sum_k x[n][k] * W_heads[head][k][d]
// Wave computes 64(M) x 64(N); block = 8 waves; grid.y = head.
// A-frag (16x4 f32): lane m=l&15, half=l>>4, reg r -> x[row0+m][k + 2*half + r]
// B-frag (4x16 f32): lane n=l&15, half,     reg r -> W[k + 2*half + r][n0 + n]
// C/D    (16x16)   : reg r, lanes 0-15 -> (M=r, N=lane); lanes 16-31 -> (M=r+8)
// ---------------------------------------------------------------------------
__global__ __launch_bounds__(256)
void spgat_gemm1_wmma(const float* __restrict__ x,
                      const float* __restrict__ W_heads,
                      float* __restrict__ hfeat, int N)
{
    const int wave = threadIdx.x >> 5;
    const int lane = threadIdx.x & 31;
    const int m    = lane & 15;
    const int half = lane >> 4;

    const int head  = blockIdx.y;
    const int mbase = (blockIdx.x * 8 + wave) * 64;   // 4 M-tiles of 16 rows
    if (mbase >= N) return;

    const float* __restrict__ Wh = W_heads + (size_t)head * (F_IN * D_HID);

    const float* __restrict__ xr0 = x + (size_t)min(mbase +  0 + m, N - 1) * F_IN + half * 2;
    const float* __restrict__ xr1 = x + (size_t)min(mbase + 16 + m, N - 1) * F_IN + half * 2;
    const float* __restrict__ xr2 = x + (size_t)min(mbase + 32 + m, N - 1) * F_IN + half * 2;
    const float* __restrict__ xr3 = x + (size_t)min(mbase + 48 + m, N - 1) * F_IN + half * 2;

    v8f acc[4][4] = {};   // [mtile][ntile]

    #pragma unroll 2
    for (int k = 0; k < F_IN; k += 4) {
        const float* __restrict__ wk = Wh + (size_t)(k + half * 2) * D_HID + m;
        v2f b0, b1, b2, b3;
        b0.x = wk[0];  b0.y = wk[D_HID + 0];
        b1.x = wk[16]; b1.y = wk[D_HID + 16];
        b2.x = wk[32]; b2.y = wk[D_HID + 32];
        b3.x = wk[48]; b3.y = wk[D_HID + 48];

        v2f a0 = *(const v2f*)(xr0 + k);
        v2f a1 = *(const v2f*)(xr1 + k);
        v2f a2 = *(const v2f*)(xr2 + k);
        v2f a3 = *(const v2f*)(xr3 + k);

        acc[0][0] = wmma_f32(a0, b0, acc[0][0]);
        acc[1][0] = wmma_f32(a1, b0, acc[1][0]);
        acc[2][0] = wmma_f32(a2, b0, acc[2][0]);
        acc[3][0] = wmma_f32(a3, b0, acc[3][0]);
        acc[0][1] = wmma_f32(a0, b1, acc[0][1]);
        acc[1][1] = wmma_f32(a1, b1, acc[1][1]);
        acc[2][1] = wmma_f32(a2, b1, acc[2][1]);
        acc[3][1] = wmma_f32(a3, b1, acc[3][1]);
        acc[0][2] = wmma_f32(a0, b2, acc[0][2]);
        acc[1][2] = wmma_f32(a1, b2, acc[1][2]);
        acc[2][2] = wmma_f32(a2, b2, acc[2][2]);
        acc[3][2] = wmma_f32(a3, b2, acc[3][2]);
        acc[0][3] = wmma_f32(a0, b3, acc[0][3]);
        acc[1][3] = wmma_f32(a1, b3, acc[1][3]);
        acc[2][3] = wmma_f32(a2, b3, acc[2][3]);
        acc[3][3] = wmma_f32(a3, b3, acc[3][3]);
    }

    #pragma unroll
    for (int mt = 0; mt < 4; ++mt) {
        const int row0 = mbase + mt * 16;
        if (row0 >= N) break;
        float* __restrict__ outp = hfeat + (size_t)row0 * HCAT + head * D_HID + m;
        #pragma unroll
        for (int r = 0; r < 8; ++r) {
            const int rr = r + 8 * half;
            if (row0 + rr < N) {
                float* orow = outp + (size_t)rr * HCAT;
                orow[0]  = acc[mt][0][r];
                orow[16] = acc[mt][1][r];
                orow[32] = acc[mt][2][r];
                orow[48] = acc[mt][3][r];
            }
        }
    }
}

// ---------------------------------------------------------------------------
// GEMM2:  h2[n][c] = sum_k xcat[n][k] * W_out[k][c]   (c padded 10 -> 16)
// Wave computes 64(M) x 16(N); one B fragment feeds 4 WMMAs.
// ---------------------------------------------------------------------------
__global__ __launch_bounds__(256)
void spgat_gemm2_wmma(const float* __restrict__ xcat,
                      const float* __restrict__ W_out,
                      float* __restrict__ h2, int N)
{
    const int wave = threadIdx.x >> 5;
    const int lane = threadIdx.x & 31;
    const int m    = lane & 15;
    const int half = lane >> 4;

    const int mbase = (blockIdx.x * 8 + wave) * 64;
    if (mbase >= N) return;

    const float* __restrict__ xr0 = xcat + (size_t)min(mbase +  0 + m, N - 1) * HCAT + half * 2;
    const float* __restrict__ xr1 = xcat + (size_t)min(mbase + 16 + m, N - 1) * HCAT + half * 2;
    const float* __restrict__ xr2 = xcat + (size_t)min(mbase + 32 + m, N - 1) * HCAT + half * 2;
    const float* __restrict__ xr3 = xcat + (size_t)min(mbase + 48 + m, N - 1) * HCAT + half * 2;
    const bool valid_n = (m < NCLS);

    v8f acc0 = {}, acc1 = {}, acc2 = {}, acc3 = {};

    #pragma unroll 4
    for (int k = 0; k < HCAT; k += 4) {
        const int kk = k + half * 2;
        v2f b;
        b.x = valid_n ? W_out[(size_t)kk       * NCLS + m] : 0.0f;
        b.y = valid_n ? W_out[(size_t)(kk + 1) * NCLS + m] : 0.0f;

        v2f a0 = *(const v2f*)(xr0 + k);
        v2f a1 = *(const v2f*)(xr1 + k);
        v2f a2 = *(const v2f*)(xr2 + k);
        v2f a3 = *(const v2f*)(xr3 + k);

        acc0 = wmma_f32(a0, b, acc0);
        acc1 = wmma_f32(a1, b, acc1);
        acc2 = wmma_f32(a2, b, acc2);
        acc3 = wmma_f32(a3, b, acc3);
    }

    v8f accs[4] = {acc0, acc1, acc2, acc3};
    #pragma unroll
    for (int mt = 0; mt < 4; ++mt) {
        const int row0 = mbase + mt * 16;
        if (row0 >= N) break;
        #pragma unroll
        for (int r = 0; r < 8; ++r) {
            const int rr = r + 8 * half;
            if (row0 + rr < N)
                h2[(size_t)(row0 + rr) * NCLSP + m] = accs[mt][r];
        }
    }
}

// ---------------------------------------------------------------------------
// Per-node attention logit halves, layer 1: s_src[n,h] = h[n,h,:]·a[h,:64],
// s_dst[n,h] = h[n,h,:]·a[h,64:]
// ---------------------------------------------------------------------------
__global__ __launch_bounds__(256)
void spgat_score1(const float* __restrict__ hfeat,
                  const float* __restrict__ a_heads,
                  float* __restrict__ s_src, float* __restrict__ s_dst, int N)
{
    const int tid = blockIdx.x * blockDim.x + threadIdx.x;
    if (tid >= N * NHEAD) return;
    const int n = tid >> 3;
    const int h = tid & 7;
    const float* __restrict__ hp = hfeat + (size_t)n * HCAT + h * D_HID;
    const float* __restrict__ ap = a_heads + (size_t)h * (2 * D_HID);
    float ss = 0.f, sd = 0.f;
    #pragma unroll 8
    for (int d = 0; d < D_HID; ++d) {
        const float v = hp[d];
        ss += v * ap[d];
        sd += v * ap[D_HID + d];
    }
    s_src[tid] = ss;
    s_dst[tid] = sd;
}

// ---------------------------------------------------------------------------
// Edge scatter, layer 1: one wave32 per (edge, head); 8 heads per block.
// ---------------------------------------------------------------------------
__global__ __launch_bounds__(256)
void spgat_edge1(const int* __restrict__ esrc, const int* __restrict__ edst,
                 const float* __restrict__ s_src, const float* __restrict__ s_dst,
                 const float* __restrict__ hfeat,
                 float* __restrict__ rowsum, float* __restrict__ agg, int E)
{
    const int e = blockIdx.x;
    if (e >= E) return;
    const int h    = threadIdx.x >> 5;
    const int lane = threadIdx.x & 31;
    const int s = esrc[e];
    const int d = edst[e];

    const float score = s_src[s * NHEAD + h] + s_dst[d * NHEAD + h];
    const float lr    = score > 0.f ? score : GAT_ALPHA * score;
    const float att   = expf(-lr);

    const float* __restrict__ hp = hfeat + (size_t)d * HCAT + h * D_HID;
    float*       __restrict__ ap = agg   + (size_t)s * HCAT + h * D_HID;
    atomicAdd(ap + lane,      att * hp[lane]);
    atomicAdd(ap + lane + 32, att * hp[lane + 32]);
    if (lane == 0) atomicAdd(rowsum + s * NHEAD + h, att);
}

// ---------------------------------------------------------------------------
// Layer-1 epilogue: xcat = elu(agg / rowsum) written in place over agg.
// ---------------------------------------------------------------------------
__global__ __launch_bounds__(256)
void spgat_finalize1(float* __restrict__ agg, const float* __restrict__ rowsum,
                     long long total)
{
    const long long i = (long long)blockIdx.x * blockDim.x + threadIdx.x;
    if (i >= total) return;
    const int n = (int)(i >> 9);
    const int j = (int)(i & 511);
    const int h = j >> 6;
    const float t = agg[i] / rowsum[n * NHEAD + h];
    agg[i] = t > 0.f ? t : expm1f(t);
}

// ---------------------------------------------------------------------------
// Per-node logit halves, layer 2.
// ---------------------------------------------------------------------------
__global__ __launch_bounds__(256)
void spgat_score2(const float* __restrict__ h2, const float* __restrict__ a_out,
                  float* __restrict__ s2s, float* __restrict__ s2d, int N)
{
    const int n = blockIdx.x * blockDim.x + threadIdx.x;
    if (n >= N) return;
    float ss = 0.f, sd = 0.f;
    #pragma unroll
    for (int c = 0; c < NCLS; ++c) {
        const float v = h2[(size_t)n * NCLSP + c];
        ss += v * a_out[c];
        sd += v * a_out[NCLS + c];
    }
    s2s[n] = ss;
    s2d[n] = sd;
}

// ---------------------------------------------------------------------------
// Edge scatter, layer 2: one wave32 per edge (lanes 0..9 carry features).
// ---------------------------------------------------------------------------
__global__ __launch_bounds__(256)
void spgat_edge2(const int* __restrict__ esrc, const int* __restrict__ edst,
                 const float* __restrict__ s2s, const float* __restrict__ s2d,
                 const float* __restrict__ h2,
                 float* __restrict__ rowsum2, float* __restrict__ agg2, int E)
{
    const int e = blockIdx.x * 8 + (threadIdx.x >> 5);
    const int lane = threadIdx.x & 31;
    if (e >= E) return;
    const int s = esrc[e];
    const int d = edst[e];
    const float score = s2s[s] + s2d[d];
    const float lr    = score > 0.f ? score : GAT_ALPHA * score;
    const float att   = expf(-lr);
    if (lane < NCLS)
        atomicAdd(agg2 + (size_t)s * NCLSP + lane, att * h2[(size_t)d * NCLSP + lane]);
    if (lane == 0)
        atomicAdd(rowsum2 + s, att);
}

// ---------------------------------------------------------------------------
// Final: out[n][c] = log_softmax( elu(agg2/rowsum2) )
// ---------------------------------------------------------------------------
__global__ __launch_bounds__(256)
void spgat_final(const float* __restrict__ agg2, const float* __restrict__ rowsum2,
                 float* __restrict__ out, int N)
{
    const int n = blockIdx.x * blockDim.x + threadIdx.x;
    if (n >= N) return;
    const float rs = rowsum2[n];
    float v[NCLS];
    float mx = -INFINITY;
    #pragma unroll
    for (int c = 0; c < NCLS; ++c) {
        float t = agg2[(size_t)n * NCLSP + c] / rs;
        t = t > 0.f ? t : expm1f(t);
        v[c] = t;
        mx = fmaxf(mx, t);
    }
    float ssum = 0.f;
    #pragma unroll
    for (int c = 0; c < NCLS; ++c) ssum += expf(v[c] - mx);
    const float lse = mx + logf(ssum);
    #pragma unroll
    for (int c = 0; c < NCLS; ++c) out[(size_t)n * NCLS + c] = v[c] - lse;
}

// ---------------------------------------------------------------------------
// Host launcher
// ---------------------------------------------------------------------------
extern "C" void kernel_launch(void* const* d_in, const int* in_sizes, int n_in,
                              void* d_out, int out_size, void* d_ws, size_t ws_size,
                              hipStream_t stream)
{
    const float* x    = (const float*)d_in[0];
    const int*   esrc = (const int*)  d_in[1];
    const int*   edst = (const int*)  d_in[2];
    const float* Wh   = (const float*)d_in[3];
    const float* ah   = (const float*)d_in[4];
    const float* Wo   = (const float*)d_in[5];
    const float* ao   = (const float*)d_in[6];
    float*       out  = (float*)d_out;

    const int N = in_sizes[0] / F_IN;
    const int E = in_sizes[1];

    // Workspace carve-out (256B aligned slices).
    char*  ws  = (char*)d_ws;
    size_t off = 0;
    auto carve = [&](size_t nfl) {
        float* p = (float*)(ws + off);
        off += ((nfl * sizeof(float) + 255) / 256) * 256;
        return p;
    };
    float* hfeat  = carve((size_t)N * HCAT);   // [N,8,64]
    float* agg    = carve((size_t)N * HCAT);   // becomes xcat in place
    float* s_src  = carve((size_t)N * NHEAD);
    float* s_dst  = carve((size_t)N * NHEAD);
    float* rowsum = carve((size_t)N * NHEAD);
    float* h2     = carve((size_t)N * NCLSP);
    float* s2s    = carve((size_t)N);
    float* s2d    = carve((size_t)N);
    float* rs2    = carve((size_t)N);
    float* agg2   = carve((size_t)N * NCLSP);

    // Zero the accumulation buffers every call (graph-capture-safe memsets).
    hipMemsetAsync(agg,    0, (size_t)N * HCAT  * sizeof(float), stream);
    hipMemsetAsync(rowsum, 0, (size_t)N * NHEAD * sizeof(float), stream);
    hipMemsetAsync(agg2,   0, (size_t)N * NCLSP * sizeof(float), stream);
    hipMemsetAsync(rs2,    0, (size_t)N         * sizeof(float), stream);

    const int wstrips = (N + 63) / 64;              // 64-row strips per wave

    dim3 g1((wstrips + 7) / 8, NHEAD);
    spgat_gemm1_wmma<<<g1, 256, 0, stream>>>(x, Wh, hfeat, N);

    spgat_score1<<<(N * NHEAD + 255) / 256, 256, 0, stream>>>(hfeat, ah, s_src, s_dst, N);

    spgat_edge1<<<E, 256, 0, stream>>>(esrc, edst, s_src, s_dst, hfeat, rowsum, agg, E);

    const long long tot1 = (long long)N * HCAT;
    spgat_finalize1<<<(unsigned)((tot1 + 255) / 256), 256, 0, stream>>>(agg, rowsum, tot1);

    spgat_gemm2_wmma<<<dim3((wstrips + 7) / 8), 256, 0, stream>>>(agg, Wo, h2, N);

    spgat_score2<<<(N + 255) / 256, 256, 0, stream>>>(h2, ao, s2s, s2d, N);

    spgat_edge2<<<(E + 7) / 8, 256, 0, stream>>>(esrc, edst, s2s, s2d, h2, rs2, agg2, E);

    spgat_final<<<(N + 255) / 256, 256, 0, stream>>>(agg2, rs2, out, N);
}